// SigMMD_55121610277609
// MI455X (gfx1250) — compile-verified
//
#include <hip/hip_runtime.h>

typedef __attribute__((ext_vector_type(2))) float v2f;
typedef __attribute__((ext_vector_type(8))) float v8f;

#define NPTS  150   // path length
#define NSEG  149   // increments per path
#define NDIM  8     // feature dim
#define TPAD  160   // padded s/t extent (10 tiles of 16)
#define NPAIR 1024  // 32*32
#define NTOT  3072  // 3 grams * 1024 pairs

// One wave (32 threads) per (gram, a, b) pair.
// Phase 1 per s-tile: inc[s0:s0+16, 0:160] = dXa * dYb^T via V_WMMA_F32_16X16X4_F32 (K=8 as 2 chained K=4).
// Phase 2: 16 steps of the Goursat recursion, t parallel across the wave via shuffle prefix scan.
__global__ __launch_bounds__(32) void sig_pde_wmma_kernel(const float* __restrict__ X,
                                                          const float* __restrict__ Y,
                                                          float* __restrict__ Kout)
{
    const int g    = blockIdx.x;          // [0, 3072)
    const int gram = g >> 10;             // 0: K(X,X), 1: K(Y,Y), 2: K(X,Y)
    const int pair = g & (NPAIR - 1);
    const int a    = pair >> 5;
    const int b    = pair & 31;

    const float* __restrict__ Pa = ((gram == 1) ? Y : X) + a * NPTS * NDIM;
    const float* __restrict__ Pb = ((gram == 0) ? X : Y) + b * NPTS * NDIM;

    __shared__ float dA[TPAD * NDIM];     // dX_a, zero-padded rows >= 149
    __shared__ float dB[TPAD * NDIM];     // dY_b, zero-padded
    __shared__ float incT[16 * TPAD];     // one 16-row strip of inc

    const int lane = threadIdx.x;

    // Build path increments in LDS (1280 elements each, 40 per lane, no divergence).
    for (int i = lane; i < TPAD * NDIM; i += 32) {
        const int s = i >> 3;
        const int d = i & 7;
        float va = 0.0f, vb = 0.0f;
        if (s < NSEG) {
            va = Pa[(s + 1) * NDIM + d] - Pa[s * NDIM + d];
            vb = Pb[(s + 1) * NDIM + d] - Pb[s * NDIM + d];
        }
        dA[i] = va;
        dB[i] = vb;
    }
    __syncthreads();

    // K row held in registers: lane owns t = 5*lane + i, i = 0..4 (lanes 30/31 padding).
    float kp[5];
#pragma unroll
    for (int i = 0; i < 5; ++i) kp[i] = 1.0f;   // K[s=0, t] = 1

    const int row  = lane & 15;    // M (or N) index inside a 16x16 tile
    const int half = lane >> 4;    // selects K {0,1} vs {2,3} per WMMA fragment layout
    const int t0v  = 5 * lane;     // first t owned by this lane

    int sGlobal = 0;
    for (int st = 0; st < 10; ++st) {
        const int s0 = st * 16;
        __syncthreads();   // incT strip from previous tile fully consumed

        // A fragment (16x4 f32): lanes 0-15 hold K=0/1, lanes 16-31 hold K=2/3.
        const int abase = (s0 + row) * NDIM + 2 * half;
        v2f afrag0, afrag1;
        afrag0.x = dA[abase + 0];
        afrag0.y = dA[abase + 1];
        afrag1.x = dA[abase + 4];
        afrag1.y = dA[abase + 5];

        for (int nt = 0; nt < 10; ++nt) {
            const int t0 = nt * 16;
            // B fragment (4x16 f32): N striped across lanes, K split like A.
            const int bbase = (t0 + row) * NDIM + 2 * half;
            v2f bfrag0, bfrag1;
            bfrag0.x = dB[bbase + 0];
            bfrag0.y = dB[bbase + 1];
            bfrag1.x = dB[bbase + 4];
            bfrag1.y = dB[bbase + 5];

            v8f acc = {};
            acc = __builtin_amdgcn_wmma_f32_16x16x4_f32(false, afrag0, false, bfrag0,
                                                        (short)0, acc, false, false);
            acc = __builtin_amdgcn_wmma_f32_16x16x4_f32(false, afrag1, false, bfrag1,
                                                        (short)0, acc, false, false);

            // C/D layout: VGPR r holds M = r + 8*half, N = t0 + row.
#pragma unroll
            for (int r = 0; r < 8; ++r)
                incT[(r + 8 * half) * TPAD + t0 + row] = acc[r];
        }
        __syncthreads();

        // 16 sequential s-steps; each is a wave-parallel update over t.
        for (int ss = 0; (ss < 16) && (sGlobal < NSEG); ++ss, ++sGlobal) {
            const float k5 = __shfl_down(kp[0], 1, 32);  // K_prev[5*(lane+1)]
            const float* irow = &incT[ss * TPAD];

            float nxt[5] = { kp[1], kp[2], kp[3], kp[4], k5 };
            float del[5];
#pragma unroll
            for (int i = 0; i < 5; ++i) {
                const int t = t0v + i;                   // delta index, valid if < 149
                const float iv = irow[t];                // t <= 159 < TPAD: always in bounds
                del[i] = (t < NSEG) ? (nxt[i] + kp[i] * (iv - 1.0f)) : 0.0f;
            }
            // Local inclusive prefix.
            float c0 = del[0];
            float c1 = c0 + del[1];
            float c2 = c1 + del[2];
            float c3 = c2 + del[3];
            float c4 = c3 + del[4];
            // Wave inclusive scan of per-lane totals (5 shuffle steps, wave32).
            float tot = c4;
#pragma unroll
            for (int off = 1; off < 32; off <<= 1) {
                const float n = __shfl_up(tot, off, 32);
                if (lane >= off) tot += n;
            }
            const float base = 1.0f + (tot - c4);        // 1 + exclusive prefix of lane totals
            kp[0] = base;                                // K_new[5*lane] (==1 for lane 0)
            kp[1] = base + c0;
            kp[2] = base + c1;
            kp[3] = base + c2;
            kp[4] = base + c3;
        }
    }

    // Final corner K[149][149]: t = 149 = 5*29 + 4 -> lane 29, kp[4].
    const float res = __shfl(kp[4], 29, 32);
    if (lane == 0) Kout[g] = res;
}

// Deterministic single-block combine: mmd = mean(KXX) + mean(KYY) - 2*mean(KXY).
__global__ __launch_bounds__(128) void sig_mmd_reduce_kernel(const float* __restrict__ K,
                                                             float* __restrict__ out)
{
    __shared__ float sm[128];
    const int tid = threadIdx.x;
    float acc = 0.0f;
    for (int i = tid; i < NTOT; i += 128) {
        const int gram = i >> 10;
        const float w = (gram == 2) ? -2.0f : 1.0f;
        acc += w * K[i];
    }
    sm[tid] = acc;
    __syncthreads();
    for (int s = 64; s > 0; s >>= 1) {
        if (tid < s) sm[tid] += sm[tid + s];
        __syncthreads();
    }
    if (tid == 0) out[0] = sm[0] * (1.0f / (float)NPAIR);
}

extern "C" void kernel_launch(void* const* d_in, const int* in_sizes, int n_in,
                              void* d_out, int out_size, void* d_ws, size_t ws_size,
                              hipStream_t stream) {
    const float* X = (const float*)d_in[0];   // (32, 150, 8) f32
    const float* Y = (const float*)d_in[1];   // (32, 150, 8) f32
    float* Kbuf = (float*)d_ws;               // 3072 floats of scratch
    float* out  = (float*)d_out;              // scalar f32

    sig_pde_wmma_kernel<<<dim3(NTOT), dim3(32), 0, stream>>>(X, Y, Kbuf);
    sig_mmd_reduce_kernel<<<dim3(1), dim3(128), 0, stream>>>(Kbuf, out);
    (void)in_sizes; (void)n_in; (void)out_size; (void)ws_size;
}